// OptimizedGraphSAGE_20512763806339
// MI455X (gfx1250) — compile-verified
//
#include <hip/hip_runtime.h>
#include <hip/hip_bf16.h>

typedef __attribute__((ext_vector_type(2))) float v2f;
typedef __attribute__((ext_vector_type(8))) float v8f;

#define D_FEAT 128   // K dimension for every GEMM (128-d features throughout)

// ---------------------------------------------------------------------------
// Degree accumulation (once per launch; graph is identical across layers)
// ---------------------------------------------------------------------------
__global__ __launch_bounds__(256)
void deg_kernel(const int* __restrict__ dst, float* __restrict__ deg, int E) {
    int e = blockIdx.x * blockDim.x + threadIdx.x;
    if (e < E) unsafeAtomicAdd(&deg[dst[e]], 1.0f);
}

__global__ __launch_bounds__(256)
void invdeg_kernel(float* __restrict__ deg, int n) {
    int i = blockIdx.x * blockDim.x + threadIdx.x;
    if (i < n) deg[i] = 1.0f / fmaxf(deg[i], 1.0f);
}

// ---------------------------------------------------------------------------
// Edge scatter-add: agg[dst] += feat[src]   (float4 gather, HW float atomics)
// one thread per (edge, 4-float group) -> 32 groups per edge
// ---------------------------------------------------------------------------
__global__ __launch_bounds__(256)
void scatter_add_kernel(const float* __restrict__ feat,
                        const int* __restrict__ src,
                        const int* __restrict__ dst,
                        float* __restrict__ agg, long long E) {
    long long gid = (long long)blockIdx.x * blockDim.x + threadIdx.x;
    if (gid >= E * 32LL) return;
    int e = (int)(gid >> 5);
    int g = (int)(gid & 31) << 2;
    int s = src[e], d = dst[e];
    const float4 v = *(const float4*)(feat + (size_t)s * D_FEAT + g);
    float* p = agg + (size_t)d * D_FEAT + g;
    unsafeAtomicAdd(p + 0, v.x);
    unsafeAtomicAdd(p + 1, v.y);
    unsafeAtomicAdd(p + 2, v.z);
    unsafeAtomicAdd(p + 3, v.w);
}

// ---------------------------------------------------------------------------
// Fused SAGE GEMM:  out = (agg * invdeg) @ Wl + feat @ Wr + b   [+ ReLU]
// One block = 16 output rows x COLS output cols. COLS/16 waves, one 16x16
// tile per wave. A tiles staged in LDS (padded stride 132 -> conflict-free
// ds_load_2addr_b64 fragment reads), WMMA fp32 16x16x4, 32 k-steps/operand.
// ---------------------------------------------------------------------------
template<int COLS>
__global__ __launch_bounds__(32 * (COLS / 16))
void sage_gemm_kernel(const float* __restrict__ agg,
                      const float* __restrict__ invdeg,
                      const float* __restrict__ feat,
                      const float* __restrict__ Wl,
                      const float* __restrict__ Wr,
                      const float* __restrict__ bias,
                      float* __restrict__ out,
                      int nrows, int do_relu) {
    constexpr int LDSS = 132;                 // padded row stride (floats)
    __shared__ float ldsA[16 * LDSS];         // mean-aggregated neighbors
    __shared__ float ldsX[16 * LDSS];         // self features

    const int r0 = blockIdx.x * 16;
    const bool fullTile = (r0 + 16 <= nrows);

    // --- coalesced float4 staging of the two 16x128 A tiles ---
    for (int f = threadIdx.x; f < 16 * 32; f += blockDim.x) {
        int row = f >> 5;
        int c4  = (f & 31) << 2;
        int gr  = r0 + row;
        if (gr < nrows) {
            float id = invdeg[gr];
            float4 a4 = *(const float4*)(agg  + (size_t)gr * D_FEAT + c4);
            float4 x4 = *(const float4*)(feat + (size_t)gr * D_FEAT + c4);
            a4.x *= id; a4.y *= id; a4.z *= id; a4.w *= id;
            *(float4*)(ldsA + row * LDSS + c4) = a4;
            *(float4*)(ldsX + row * LDSS + c4) = x4;
        }
    }
    __syncthreads();

    const int wave = threadIdx.x >> 5;
    const int lane = threadIdx.x & 31;
    const int ln   = lane & 15;
    const bool hi  = lane >= 16;
    const int c0   = wave * 16;

    // accumulator seeded with bias (broadcast down rows)
    v8f acc;
    {
        float bv = bias[c0 + ln];
        #pragma unroll
        for (int i = 0; i < 8; ++i) acc[i] = bv;
    }

    // pass 1: mean-agg @ Wl ; pass 2: self @ Wr
    #pragma unroll 1
    for (int pass = 0; pass < 2; ++pass) {
        const float* A = pass == 0 ? ldsA : ldsX;
        const float* W = pass == 0 ? Wl : Wr;
        #pragma unroll 8
        for (int kk = 0; kk < D_FEAT; kk += 4) {
            const int kA = kk + (hi ? 2 : 0);
            v2f a, b;
            a.x = A[ln * LDSS + kA];
            a.y = A[ln * LDSS + kA + 1];
            b.x = W[(size_t)kA * COLS + c0 + ln];
            b.y = W[(size_t)(kA + 1) * COLS + c0 + ln];
            acc = __builtin_amdgcn_wmma_f32_16x16x4_f32(
                false, a, false, b, (short)0, acc, false, false);
        }
    }

    // C/D layout: VGPR i -> row i (lanes 0-15) / row i+8 (lanes 16-31)
    const int rbase = r0 + (hi ? 8 : 0);
    if (fullTile) {
        // uniform fast path: no per-element exec masking, stores can clause
        if (do_relu) {
            #pragma unroll
            for (int i = 0; i < 8; ++i)
                out[(size_t)(rbase + i) * COLS + c0 + ln] = fmaxf(acc[i], 0.0f);
        } else {
            #pragma unroll
            for (int i = 0; i < 8; ++i)
                out[(size_t)(rbase + i) * COLS + c0 + ln] = acc[i];
        }
    } else {
        #pragma unroll
        for (int i = 0; i < 8; ++i) {
            int r = rbase + i;
            if (r < nrows) {
                float v = acc[i];
                if (do_relu) v = fmaxf(v, 0.0f);
                out[(size_t)r * COLS + c0 + ln] = v;
            }
        }
    }
}

// ---------------------------------------------------------------------------
// BatchNorm column statistics: stats[c] = mu, stats[128+c] = rsqrt(var+eps)
// (h1 is L2-resident at 25.6 MB vs 192 MB L2, so the column-strided reads
//  cost HBM bandwidth only once)
// ---------------------------------------------------------------------------
__global__ __launch_bounds__(256)
void bn_stats_kernel(const float* __restrict__ h, float* __restrict__ stats, int n) {
    const int c = blockIdx.x;  // 0..127
    float s1 = 0.0f, s2 = 0.0f;
    for (int r = threadIdx.x; r < n; r += blockDim.x) {
        float v = h[(size_t)r * D_FEAT + c];
        s1 += v; s2 += v * v;
    }
    __shared__ float a1[256], a2[256];
    a1[threadIdx.x] = s1; a2[threadIdx.x] = s2;
    __syncthreads();
    for (int s = 128; s > 0; s >>= 1) {
        if (threadIdx.x < s) {
            a1[threadIdx.x] += a1[threadIdx.x + s];
            a2[threadIdx.x] += a2[threadIdx.x + s];
        }
        __syncthreads();
    }
    if (threadIdx.x == 0) {
        float mu  = a1[0] / (float)n;
        float var = a2[0] / (float)n - mu * mu;
        stats[c]            = mu;
        stats[D_FEAT + c]   = rsqrtf(var + 1e-5f);
    }
}

// ---------------------------------------------------------------------------
// h = relu(BN(h1)) + x  (elementwise over N x 128)
// ---------------------------------------------------------------------------
__global__ __launch_bounds__(256)
void bn_relu_res_kernel(const float* __restrict__ h1,
                        const float* __restrict__ x,
                        const float* __restrict__ stats,
                        const float* __restrict__ gamma,
                        const float* __restrict__ beta,
                        float* __restrict__ out, long long total) {
    long long i = (long long)blockIdx.x * blockDim.x + threadIdx.x;
    if (i >= total) return;
    int c = (int)(i & (D_FEAT - 1));
    float v = (h1[i] - stats[c]) * stats[D_FEAT + c] * gamma[c] + beta[c];
    out[i] = fmaxf(v, 0.0f) + x[i];
}

// ---------------------------------------------------------------------------
// In-place log_softmax over 64 cols; one wave32 per row (2 cols/lane)
// ---------------------------------------------------------------------------
__global__ __launch_bounds__(256)
void log_softmax64_kernel(float* __restrict__ out, int nrows) {
    int row = blockIdx.x * 8 + (threadIdx.x >> 5);
    if (row >= nrows) return;
    int lane = threadIdx.x & 31;
    float* p = out + (size_t)row * 64;
    float v0 = p[lane], v1 = p[lane + 32];
    float m = fmaxf(v0, v1);
    #pragma unroll
    for (int off = 16; off > 0; off >>= 1) m = fmaxf(m, __shfl_xor(m, off, 32));
    float s = expf(v0 - m) + expf(v1 - m);
    #pragma unroll
    for (int off = 16; off > 0; off >>= 1) s += __shfl_xor(s, off, 32);
    float l = m + logf(s);
    p[lane]      = v0 - l;
    p[lane + 32] = v1 - l;
}

// ---------------------------------------------------------------------------
// Launch: inputs in setup_inputs() order:
// 0:x 1:edge_index 2:Wl0 3:Wr0 4:b0 5:Wl1 6:Wr1 7:b1 8:Wl2 9:Wr2 10:b2
// 11:gamma 12:beta
// ---------------------------------------------------------------------------
extern "C" void kernel_launch(void* const* d_in, const int* in_sizes, int n_in,
                              void* d_out, int out_size, void* d_ws, size_t ws_size,
                              hipStream_t stream) {
    const float* x    = (const float*)d_in[0];
    const int*   ei   = (const int*)d_in[1];
    const float* Wl0  = (const float*)d_in[2];
    const float* Wr0  = (const float*)d_in[3];
    const float* b0   = (const float*)d_in[4];
    const float* Wl1  = (const float*)d_in[5];
    const float* Wr1  = (const float*)d_in[6];
    const float* b1   = (const float*)d_in[7];
    const float* Wl2  = (const float*)d_in[8];
    const float* Wr2  = (const float*)d_in[9];
    const float* b2   = (const float*)d_in[10];
    const float* gamma= (const float*)d_in[11];
    const float* beta = (const float*)d_in[12];

    const int N = in_sizes[0] / D_FEAT;     // 50000
    const int E = in_sizes[1] / 2;          // 600000
    const int* src = ei;
    const int* dst = ei + E;

    // workspace layout
    char* ws = (char*)d_ws;
    const size_t featBytes = (size_t)N * D_FEAT * sizeof(float);
    float* agg    = (float*)ws;                              ws += featBytes;
    float* bufA   = (float*)ws;                              ws += featBytes; // h0, then h_mid
    float* bufB   = (float*)ws;                              ws += featBytes; // h1
    float* invdeg = (float*)ws;                              ws += (size_t)N * sizeof(float);
    float* stats  = (float*)ws;                              // 256 floats

    float* out = (float*)d_out;             // N x 64

    const int edgeBlocks    = (E + 255) / 256;
    const long long sc      = (long long)E * 32LL;
    const int scatterBlocks = (int)((sc + 255) / 256);
    const int rowTiles      = (N + 15) / 16;
    const long long elems   = (long long)N * D_FEAT;
    const int ewBlocks      = (int)((elems + 255) / 256);

    // degree (once)
    hipMemsetAsync(invdeg, 0, (size_t)N * sizeof(float), stream);
    deg_kernel<<<edgeBlocks, 256, 0, stream>>>(dst, invdeg, E);
    invdeg_kernel<<<(N + 255) / 256, 256, 0, stream>>>(invdeg, N);

    // ---- layer 0: h0 = relu(mean(x) @ Wl0 + x @ Wr0 + b0) ----
    hipMemsetAsync(agg, 0, featBytes, stream);
    scatter_add_kernel<<<scatterBlocks, 256, 0, stream>>>(x, src, dst, agg, E);
    sage_gemm_kernel<128><<<rowTiles, 256, 0, stream>>>(agg, invdeg, x, Wl0, Wr0, b0,
                                                        bufA, N, 1);

    // ---- layer 1: h1 = mean(h0) @ Wl1 + h0 @ Wr1 + b1 ----
    hipMemsetAsync(agg, 0, featBytes, stream);
    scatter_add_kernel<<<scatterBlocks, 256, 0, stream>>>(bufA, src, dst, agg, E);
    sage_gemm_kernel<128><<<rowTiles, 256, 0, stream>>>(agg, invdeg, bufA, Wl1, Wr1, b1,
                                                        bufB, N, 0);

    // ---- BN + relu + residual: h_mid = relu(BN(h1)) + x  (into bufA) ----
    bn_stats_kernel<<<D_FEAT, 256, 0, stream>>>(bufB, stats, N);
    bn_relu_res_kernel<<<ewBlocks, 256, 0, stream>>>(bufB, x, stats, gamma, beta,
                                                     bufA, elems);

    // ---- layer 2: out = mean(h_mid) @ Wl2 + h_mid @ Wr2 + b2 ----
    hipMemsetAsync(agg, 0, featBytes, stream);
    scatter_add_kernel<<<scatterBlocks, 256, 0, stream>>>(bufA, src, dst, agg, E);
    sage_gemm_kernel<64><<<rowTiles, 128, 0, stream>>>(agg, invdeg, bufA, Wl2, Wr2, b2,
                                                       out, N, 0);

    // ---- log_softmax over 64 classes, in place ----
    log_softmax64_kernel<<<(N + 7) / 8, 256, 0, stream>>>(out, N);
}